// SplineActivation_20014547599871
// MI455X (gfx1250) — compile-verified
//
#include <hip/hip_runtime.h>
#include <stdint.h>

// ---------------------------------------------------------------------------
// SplineActivation for MI455X (gfx1250).
// Memory-bound elementwise op: 16.78M fp32 in, 16.78M fp32 out (~134 MB
// traffic -> ~5.8us floor at 23.3 TB/s). Cox-de Boor over uniform knots is
// collapsed into 13 per-interval cubic polynomials held in LDS; per element:
// clamp, interval find, one ds_load_b128, 3-FMA Horner, edge select.
// Coeffs are staged into LDS with the CDNA5 Tensor Data Mover
// (tensor_load_to_lds + s_wait_tensorcnt), x is streamed with b128 loads +
// global_prefetch.
// ---------------------------------------------------------------------------

#define NSPL 10           // NUM_SPLINES
#define NINT 13           // knot intervals: NUM_KNOTS-1 = 13
#define INV_H 3.25f       // 13/4, exact in fp32

typedef unsigned int uint32;
typedef uint32 u32x4 __attribute__((ext_vector_type(4)));
typedef int    i32x8 __attribute__((ext_vector_type(8)));
typedef int    i32x4 __attribute__((ext_vector_type(4)));

__device__ __forceinline__ float spline_eval(float xi, const float4* tab, float c9) {
    float xc = fminf(fmaxf(xi, -2.0f), 2.0f);
    float u  = (xc + 2.0f) * INV_H;           // in [0, 13]
    int   j  = (int)u;                        // u >= 0 -> truncation == floor
    j = (j > NINT - 1) ? (NINT - 1) : j;
    float t  = u - (float)j;
    float4 cf = tab[j];                       // ds_load_b128, 16B aligned
    float r = fmaf(fmaf(fmaf(cf.w, t, cf.z), t, cf.y), t, cf.x);
    // reference: basis[...,9] += (xc == knots[-1]); all other basis are 0 there
    return (xc == 2.0f) ? c9 : r;
}

__global__ void __launch_bounds__(256)
spline_activation_kernel(const float* __restrict__ x,
                         const float* __restrict__ coeffs,
                         float* __restrict__ out,
                         int n4, int n) {
    // padded coeff vector: cpad[i] = coeffs[i-3] for i in [3,12], else 0
    __shared__ float  lds_c[16];
    __shared__ float4 lds_tab[NINT];

    const int tid = threadIdx.x;

    // zero only the pad entries (TDM owns [3..12] -> no write race)
    if (tid < 3)               lds_c[tid] = 0.0f;
    if (tid >= 13 && tid < 16) lds_c[tid] = 0.0f;

#if defined(__gfx1250__) && __has_builtin(__builtin_amdgcn_tensor_load_to_lds) && \
    __has_builtin(__builtin_amdgcn_s_wait_tensorcnt)
    if (tid == 0) {
        // Tensor DMA descriptor (CDNA5 ISA): 1D, 10 x 4B elements.
        uint32 lds_addr = (uint32)(uintptr_t)&lds_c[3];     // flat LDS ptr low 32b = LDS offset
        unsigned long long ga = (unsigned long long)(uintptr_t)coeffs;
        u32x4 g0;
        g0[0] = 1u;                                         // count=1, user descriptor
        g0[1] = lds_addr;                                   // lds_addr
        g0[2] = (uint32)(ga & 0xFFFFFFFFu);                 // global_addr[31:0]
        g0[3] = (uint32)((ga >> 32) & 0x1FFFFFFu)           // global_addr[56:32]
              | (2u << 30);                                 // type = 2 ("image")
        i32x8 g1;
        g1[0] = 0x00020000;                                 // workgroup_mask=0, data_size=2 (4B)
        g1[1] = (NSPL << 16);                               // tensor_dim0 = 10
        g1[2] = (1 << 16);                                  // tensor_dim1 = 1
        g1[3] = (NSPL << 16);                               // tile_dim0 = 10
        g1[4] = 1;                                          // tile_dim1 = 1
        g1[5] = NSPL;                                       // tensor_dim0_stride = 10
        g1[6] = 0;
        g1[7] = 0;
        i32x4 gz4 = {0, 0, 0, 0};                           // groups 2/3 unused (<=2D tensor)
        i32x8 gz8 = {0, 0, 0, 0, 0, 0, 0, 0};
        // amdgpu-toolchain (clang-23) 6-arg form
        __builtin_amdgcn_tensor_load_to_lds(g0, g1, gz4, gz4, gz8, 0);
        __builtin_amdgcn_s_wait_tensorcnt(0);               // TDM done before barrier signal
    }
#else
    if (tid < NSPL) lds_c[3 + tid] = coeffs[tid];
#endif
    __syncthreads();

    // Per-interval cubic: out|interval j = a + b t + c t^2 + d t^3, t=(x-knot_j)/h.
    // Uniform B-spline blending weights folded with the padded coeffs.
    if (tid < NINT) {
        float P = lds_c[tid], Q = lds_c[tid + 1], R = lds_c[tid + 2], S = lds_c[tid + 3];
        float4 v;
        v.x = (P + 4.0f * Q + R) * (1.0f / 6.0f);
        v.y = (R - P) * 0.5f;
        v.z = (P - 2.0f * Q + R) * 0.5f;
        v.w = (S - P + 3.0f * (Q - R)) * (1.0f / 6.0f);
        lds_tab[tid] = v;
    }
    __syncthreads();

    const float c9 = coeffs[NSPL - 1];        // uniform -> s_load, L2-resident

    const float4* __restrict__ xv = (const float4*)x;
    float4* __restrict__ ov = (float4*)out;
    const int stride = (int)(gridDim.x * blockDim.x);

    for (int i = (int)(blockIdx.x * blockDim.x) + tid; i < n4; i += stride) {
        __builtin_prefetch((const void*)(xv + i + stride), 0, 0);  // global_prefetch_b8
        float4 xin = xv[i];                                        // global_load_b128
        float4 r;
        r.x = spline_eval(xin.x, lds_tab, c9);
        r.y = spline_eval(xin.y, lds_tab, c9);
        r.z = spline_eval(xin.z, lds_tab, c9);
        r.w = spline_eval(xin.w, lds_tab, c9);
        ov[i] = r;                                                 // global_store_b128
    }

    // scalar tail (n not divisible by 4; not hit for the reference shape)
    for (int i = n4 * 4 + (int)(blockIdx.x * blockDim.x) + tid; i < n; i += stride) {
        out[i] = spline_eval(x[i], lds_tab, c9);
    }
}

extern "C" void kernel_launch(void* const* d_in, const int* in_sizes, int n_in,
                              void* d_out, int out_size, void* d_ws, size_t ws_size,
                              hipStream_t stream) {
    (void)n_in; (void)out_size; (void)d_ws; (void)ws_size;
    const float* x      = (const float*)d_in[0];
    const float* coeffs = (const float*)d_in[1];
    float* out          = (float*)d_out;

    const int n  = in_sizes[0];               // 32*128*64*64 = 16,777,216
    const int n4 = n >> 2;

    const int threads = 256;                  // 8 wave32s per block
    long long work = (n4 > 0) ? (long long)n4 : (long long)n;
    int blocks = (int)((work + threads - 1) / threads);
    if (blocks > 8192) blocks = 8192;         // grid-stride; >64K waves in flight
    if (blocks < 1)    blocks = 1;

    spline_activation_kernel<<<blocks, threads, 0, stream>>>(x, coeffs, out, n4, n);
}